// GCNLayer_35158602285513
// MI455X (gfx1250) — compile-verified
//
#include <hip/hip_runtime.h>

typedef __attribute__((ext_vector_type(2))) float v2f;
typedef __attribute__((ext_vector_type(8))) float v8f;

#define GCN_D 128  // D_IN == D_OUT == 128

// ---------------- K1: zero workspace (agg, deg, s) ----------------
__global__ void gcn_zero_ws(float* __restrict__ agg, int* __restrict__ deg,
                            float* __restrict__ s, int n_nodes) {
    const int tid = blockIdx.x * blockDim.x + threadIdx.x;
    const int stride = gridDim.x * blockDim.x;
    const long long nfeat = (long long)n_nodes * GCN_D;
    for (long long i = tid; i < nfeat; i += stride) agg[i] = 0.0f;
    for (int i = tid; i < n_nodes; i += stride) { deg[i] = 0; s[i] = 0.0f; }
}

// ---------------- K2: degree count ----------------
__global__ void gcn_degree(const int* __restrict__ row, int* __restrict__ deg,
                           int n_edges) {
    const int tid = blockIdx.x * blockDim.x + threadIdx.x;
    const int stride = gridDim.x * blockDim.x;
    for (int e = tid; e < n_edges; e += stride) {
        atomicAdd(&deg[row[e]], 1);
    }
}

// ---------------- K3: deg^{-1/2} ----------------
__global__ void gcn_dis(const int* __restrict__ deg, float* __restrict__ dis,
                        int n_nodes) {
    const int i = blockIdx.x * blockDim.x + threadIdx.x;
    if (i < n_nodes) {
        const int d = deg[i];
        dis[i] = (d > 0) ? rsqrtf((float)d) : 0.0f;
    }
}

// ---------------- K4: edge aggregation in input space ----------------
// agg[row] += norm * x[col]  (one wave of 32 lanes per edge, float4 per lane)
// s[row]   += norm           (lane 0)
__global__ void __launch_bounds__(256)
gcn_aggregate(const float* __restrict__ x, const int* __restrict__ row,
              const int* __restrict__ col, const float* __restrict__ dis,
              float* __restrict__ agg, float* __restrict__ s, int n_edges) {
    const int lane   = threadIdx.x & 31;
    const int wave   = (blockIdx.x * blockDim.x + threadIdx.x) >> 5;
    const int nwaves = (gridDim.x * blockDim.x) >> 5;

    for (int e = wave; e < n_edges; e += nwaves) {
        if (lane == 0) {
            // prefetch next index pair (emits global_prefetch_b8)
            __builtin_prefetch(row + e + nwaves, 0, 0);
            __builtin_prefetch(col + e + nwaves, 0, 0);
        }
        const int r = row[e];
        const int c = col[e];
        const float norm = dis[r] * dis[c];

        const float4 v = ((const float4*)(x + (size_t)c * GCN_D))[lane];
        float* ao = agg + (size_t)r * GCN_D + lane * 4;
        atomicAdd(ao + 0, norm * v.x);
        atomicAdd(ao + 1, norm * v.y);
        atomicAdd(ao + 2, norm * v.z);
        atomicAdd(ao + 3, norm * v.w);
        if (lane == 0) atomicAdd(&s[r], norm);
    }
}

// ---------------- K5: WMMA GEMM epilogue ----------------
// out[m][n] = sum_k agg[m][k] * W[n][k]  +  s[m] * bias[n]
// One wave computes one 16x16 output tile via V_WMMA_F32_16X16X4_F32, K=128.
__global__ void __launch_bounds__(256)
gcn_gemm_wmma(const float* __restrict__ agg, const float* __restrict__ W,
              const float* __restrict__ bias, const float* __restrict__ s,
              float* __restrict__ out) {
    const int lane = threadIdx.x & 31;
    const int nt   = threadIdx.x >> 5;     // wave-in-block = N tile (0..7)
    const int m0   = blockIdx.x * 16;
    const int n0   = nt * 16;
    const int half = lane >> 4;            // 0: lanes 0-15, 1: lanes 16-31
    const int r    = lane & 15;

    // A frag: row m0+r of agg; per step K = k0 + 2*half + {0,1}  (float2)
    const float* arow = agg + (size_t)(m0 + r) * GCN_D;
    // B frag: B[k][n] = W[n][k]; lane holds column n0+r -> row n0+r of W (float2)
    const float* wrow = W + (size_t)(n0 + r) * GCN_D;

    v8f c = {};
#pragma unroll 8
    for (int k0 = 0; k0 < GCN_D; k0 += 4) {
        const v2f a = *(const v2f*)(arow + k0 + 2 * half);
        const v2f b = *(const v2f*)(wrow + k0 + 2 * half);
        // D = A(16x4) * B(4x16) + C   -- full fp32 precision
        c = __builtin_amdgcn_wmma_f32_16x16x4_f32(
                /*neg_a=*/false, a, /*neg_b=*/false, b,
                /*c_mod=*/(short)0, c, /*reuse_a=*/false, /*reuse_b=*/false);
    }

    // Epilogue: C/D layout => element v maps to (M = v + 8*half, N = r)
    const float bn = bias[n0 + r];
#pragma unroll
    for (int v = 0; v < 8; ++v) {
        const int m = m0 + v + 8 * half;
        out[(size_t)m * GCN_D + n0 + r] = c[v] + s[m] * bn;
    }
}

extern "C" void kernel_launch(void* const* d_in, const int* in_sizes, int n_in,
                              void* d_out, int out_size, void* d_ws, size_t ws_size,
                              hipStream_t stream) {
    const float* x    = (const float*)d_in[0];
    const int*   ei   = (const int*)d_in[1];   // [2, E] flattened
    const float* W    = (const float*)d_in[2]; // [128, 128]
    const float* bias = (const float*)d_in[3]; // [128]
    float* out = (float*)d_out;

    const int n_nodes = in_sizes[0] / GCN_D;   // 100000
    const int n_edges = in_sizes[1] / 2;       // 3200000
    const int* row = ei;
    const int* col = ei + n_edges;

    // Workspace layout (packed back to back, no alignment padding):
    //   agg: n_nodes*128 f32 (51.2 MB) | deg: n_nodes i32 | dis: n_nodes f32 | s: n_nodes f32
    char* ws = (char*)d_ws;
    const size_t agg_bytes = (size_t)n_nodes * GCN_D * sizeof(float);
    float* agg = (float*)ws;
    int*   deg = (int*)(ws + agg_bytes);
    float* dis = (float*)(ws + agg_bytes + (size_t)n_nodes * 4);
    float* s   = (float*)(ws + agg_bytes + (size_t)n_nodes * 8);

    gcn_zero_ws<<<2048, 256, 0, stream>>>(agg, deg, s, n_nodes);
    gcn_degree<<<2048, 256, 0, stream>>>(row, deg, n_edges);
    gcn_dis<<<(n_nodes + 255) / 256, 256, 0, stream>>>(deg, dis, n_nodes);
    gcn_aggregate<<<8192, 256, 0, stream>>>(x, row, col, dis, agg, s, n_edges);
    gcn_gemm_wmma<<<n_nodes / 16, 256, 0, stream>>>(agg, W, bias, s, out);
}